// GraphGNN_73332271612087
// MI455X (gfx1250) — compile-verified
//
#include <hip/hip_runtime.h>
#include <hip/hip_bf16.h>

typedef __attribute__((ext_vector_type(2))) float v2f;
typedef __attribute__((ext_vector_type(8))) float v8f;

// ---------------------------------------------------------------------------
// Degree accumulation: deg[dst[e]] += 1
// ---------------------------------------------------------------------------
__global__ void deg_kernel(const int* __restrict__ dst, float* __restrict__ deg, int E) {
    int e = blockIdx.x * blockDim.x + threadIdx.x;
    if (e < E) atomicAdd(&deg[dst[e]], 1.0f);
}

// dinv[i] = rsqrt(deg[i] + 1)   (in place)
__global__ void dinv_kernel(float* __restrict__ deg, int n) {
    int i = blockIdx.x * blockDim.x + threadIdx.x;
    if (i < n) deg[i] = rsqrtf(deg[i] + 1.0f);
}

// ---------------------------------------------------------------------------
// WMMA fp32 GEMM:  H[M x N] = X[M x K] * W[K x N]   (row major, K==128)
// Block = 256 threads = 8 waves. Block computes 128 rows x 16 cols.
// Wave computes one 16x16 tile via V_WMMA_F32_16X16X4_F32, K-step 4.
// ---------------------------------------------------------------------------
__global__ __launch_bounds__(256) void gemm_wmma_f32(
    const float* __restrict__ X, const float* __restrict__ W,
    float* __restrict__ H, int M, int K, int N)
{
    __shared__ float sW[128 * 16];          // W column panel [K x 16]

    int nTiles = N >> 4;
    int nT = blockIdx.x % nTiles;
    int mG = blockIdx.x / nTiles;

    // cooperative load of W[:, nT*16 .. nT*16+15] into LDS
    for (int i = threadIdx.x; i < K * 16; i += 256) {
        int k = i >> 4, n = i & 15;
        sW[i] = W[(size_t)k * N + (nT << 4) + n];
    }
    __syncthreads();

    int wave = threadIdx.x >> 5;
    int lane = threadIdx.x & 31;
    int mBase = ((mG << 3) + wave) << 4;    // this wave's 16-row tile
    if (mBase < M) {
        int nLoc  = lane & 15;
        int kOff  = (lane >> 4) << 1;       // 0 for lanes 0-15, 2 for lanes 16-31
        const float* xrow = X + (size_t)(mBase + nLoc) * K + kOff;

        v8f acc = {};
        for (int k = 0; k < K; k += 4) {
            // A fragment (16x4): lane<16 holds K=k,k+1 ; lane>=16 holds K=k+2,k+3
            v2f a;
            a.x = xrow[k];
            a.y = xrow[k + 1];
            // B fragment (4x16): mirrored K layout, N striped over lanes
            v2f b;
            b.x = sW[(k + kOff) * 16 + nLoc];
            b.y = sW[(k + kOff + 1) * 16 + nLoc];
            acc = __builtin_amdgcn_wmma_f32_16x16x4_f32(
                false, a, false, b, (short)0, acc, false, false);
        }

        // D layout: VGPR r -> row r (lanes 0-15) / row r+8 (lanes 16-31)
        int col = (nT << 4) + nLoc;
        int r0  = mBase + ((lane >> 4) << 3);
        float* hp = H + (size_t)r0 * N + col;
#pragma unroll
        for (int r = 0; r < 8; ++r)
            hp[(size_t)r * N] = acc[r];
    }
}

// ---------------------------------------------------------------------------
// Edge aggregation: agg[dst] += dinv[src]*dinv[dst] * h[src]
// One wave per edge; each lane covers 4 contiguous features.
// ---------------------------------------------------------------------------
__global__ __launch_bounds__(256) void agg_edges(
    const float* __restrict__ h, float* __restrict__ agg,
    const int* __restrict__ src, const int* __restrict__ dst,
    const float* __restrict__ dinv, int E, int D)
{
    int gtid = blockIdx.x * blockDim.x + threadIdx.x;
    int e    = gtid >> 5;
    int lane = gtid & 31;
    if (e >= E) return;
    int s = src[e], d = dst[e];
    float norm = dinv[s] * dinv[d];
    int c = lane << 2;
    if (c < D) {
        float4 v = *(const float4*)(h + (size_t)s * D + c);
        float* p = agg + (size_t)d * D + c;
        atomicAdd(p + 0, norm * v.x);
        atomicAdd(p + 1, norm * v.y);
        atomicAdd(p + 2, norm * v.z);
        atomicAdd(p + 3, norm * v.w);
    }
}

// ---------------------------------------------------------------------------
// Epilogue: agg[i,c] = act(agg[i,c] + dinv[i]^2 * h[i,c] + b[c])
// ---------------------------------------------------------------------------
__global__ void epilogue_kernel(
    float* __restrict__ agg, const float* __restrict__ h,
    const float* __restrict__ dinv, const float* __restrict__ bias,
    int n, int D, int doRelu)
{
    size_t idx = (size_t)blockIdx.x * blockDim.x + threadIdx.x;
    if (idx >= (size_t)n * D) return;
    int i = (int)(idx / D);
    int c = (int)(idx - (size_t)i * D);
    float di = dinv[i];
    float v = agg[idx] + di * di * h[idx] + bias[c];
    if (doRelu) v = fmaxf(v, 0.0f);
    agg[idx] = v;
}

// ---------------------------------------------------------------------------
// Global mean pool stage 1: per-block LDS [64 graphs x 64 feat] accumulator,
// flushed with global atomics. Threads: c = tid&63 (feature), slot = tid>>6.
// ---------------------------------------------------------------------------
__global__ __launch_bounds__(256) void pool_kernel(
    const float* __restrict__ h, const int* __restrict__ batch,
    float* __restrict__ out, float* __restrict__ counts, int nNodes)
{
    __shared__ float sSum[64 * 64];
    __shared__ float sCnt[64];
    for (int i = threadIdx.x; i < 64 * 64; i += 256) sSum[i] = 0.0f;
    if (threadIdx.x < 64) sCnt[threadIdx.x] = 0.0f;
    __syncthreads();

    int c    = threadIdx.x & 63;
    int slot = threadIdx.x >> 6;          // 0..3
    int base = blockIdx.x * 256;
    for (int n = slot; n < 256; n += 4) {
        int node = base + n;
        if (node < nNodes) {
            int g = batch[node];
            atomicAdd(&sSum[g * 64 + c], h[(size_t)node * 64 + c]);
            if (c == 0) atomicAdd(&sCnt[g], 1.0f);
        }
    }
    __syncthreads();

    for (int i = threadIdx.x; i < 64 * 64; i += 256) {
        float v = sSum[i];
        if (v != 0.0f) atomicAdd(&out[i], v);
    }
    if (threadIdx.x < 64) {
        float v = sCnt[threadIdx.x];
        if (v != 0.0f) atomicAdd(&counts[threadIdx.x], v);
    }
}

// out[g,c] /= max(count[g], 1)
__global__ void pool_div_kernel(float* __restrict__ out, const float* __restrict__ counts) {
    int idx = blockIdx.x * blockDim.x + threadIdx.x;
    if (idx < 64 * 64) out[idx] /= fmaxf(counts[idx >> 6], 1.0f);
}

// ---------------------------------------------------------------------------
extern "C" void kernel_launch(void* const* d_in, const int* in_sizes, int n_in,
                              void* d_out, int out_size, void* d_ws, size_t ws_size,
                              hipStream_t stream)
{
    const float* x     = (const float*)d_in[0];
    const int*   ei    = (const int*)  d_in[1];
    const int*   batch = (const int*)  d_in[2];
    const float* W1    = (const float*)d_in[3];
    const float* b1    = (const float*)d_in[4];
    const float* W2    = (const float*)d_in[5];
    const float* b2    = (const float*)d_in[6];
    const float* W3    = (const float*)d_in[7];
    const float* b3    = (const float*)d_in[8];
    float*       out   = (float*)d_out;

    const int D_IN = 128, D_HID = 128, D_OUT = 64;
    int nNodes = in_sizes[0] / D_IN;        // 50000
    int E      = in_sizes[1] / 2;           // 800000
    const int* src = ei;
    const int* dst = ei + E;

    float* dinv   = (float*)d_ws;                       // nNodes
    float* counts = dinv + nNodes;                      // 64
    float* bufA   = counts + 64;                        // nNodes*128
    float* bufB   = bufA + (size_t)nNodes * D_HID;      // nNodes*128

    // ---- degrees / normalization ----
    hipMemsetAsync(dinv, 0, (size_t)nNodes * sizeof(float), stream);
    deg_kernel<<<(E + 255) / 256, 256, 0, stream>>>(dst, dinv, E);
    dinv_kernel<<<(nNodes + 255) / 256, 256, 0, stream>>>(dinv, nNodes);

    int mTiles  = (nNodes + 15) / 16;       // 3125
    int mGroups = (mTiles + 7) / 8;         // 391
    int aggBlocks = (int)(((size_t)E * 32 + 255) / 256);

    // ---- layer 1: h = relu(gcn(x, W1, b1)) -> bufB ----
    gemm_wmma_f32<<<mGroups * (D_HID / 16), 256, 0, stream>>>(x, W1, bufA, nNodes, D_IN, D_HID);
    hipMemsetAsync(bufB, 0, (size_t)nNodes * D_HID * sizeof(float), stream);
    agg_edges<<<aggBlocks, 256, 0, stream>>>(bufA, bufB, src, dst, dinv, E, D_HID);
    epilogue_kernel<<<(int)(((size_t)nNodes * D_HID + 255) / 256), 256, 0, stream>>>(
        bufB, bufA, dinv, b1, nNodes, D_HID, 1);

    // ---- layer 2: h = relu(gcn(bufB, W2, b2)) -> bufB ----
    gemm_wmma_f32<<<mGroups * (D_HID / 16), 256, 0, stream>>>(bufB, W2, bufA, nNodes, D_HID, D_HID);
    hipMemsetAsync(bufB, 0, (size_t)nNodes * D_HID * sizeof(float), stream);
    agg_edges<<<aggBlocks, 256, 0, stream>>>(bufA, bufB, src, dst, dinv, E, D_HID);
    epilogue_kernel<<<(int)(((size_t)nNodes * D_HID + 255) / 256), 256, 0, stream>>>(
        bufB, bufA, dinv, b2, nNodes, D_HID, 1);

    // ---- layer 3: h = gcn(bufB, W3, b3) -> bufB (N x 64) ----
    gemm_wmma_f32<<<mGroups * (D_OUT / 16), 256, 0, stream>>>(bufB, W3, bufA, nNodes, D_HID, D_OUT);
    hipMemsetAsync(bufB, 0, (size_t)nNodes * D_OUT * sizeof(float), stream);
    agg_edges<<<aggBlocks, 256, 0, stream>>>(bufA, bufB, src, dst, dinv, E, D_OUT);
    epilogue_kernel<<<(int)(((size_t)nNodes * D_OUT + 255) / 256), 256, 0, stream>>>(
        bufB, bufA, dinv, b3, nNodes, D_OUT, 0);

    // ---- global mean pool -> d_out (64 x 64) ----
    hipMemsetAsync(out, 0, 64 * 64 * sizeof(float), stream);
    hipMemsetAsync(counts, 0, 64 * sizeof(float), stream);
    pool_kernel<<<(nNodes + 255) / 256, 256, 0, stream>>>(bufB, batch, out, counts, nNodes);
    pool_div_kernel<<<16, 256, 0, stream>>>(out, counts);
}